// TimeSeriesWineSNN_16655883173888
// MI455X (gfx1250) — compile-verified
//
#include <hip/hip_runtime.h>

typedef __attribute__((ext_vector_type(2))) float v2f;
typedef __attribute__((ext_vector_type(4))) float v4f;
typedef __attribute__((ext_vector_type(8))) float v8f;

#define TT 1000
#define BB 4096

// ---- CDNA5 WMMA: D[16x16 f32] = A[16x4 f32] * B[4x16 f32] + C ----
__device__ __forceinline__ v8f wmma4(v2f a, v2f b, v8f c) {
  return __builtin_amdgcn_wmma_f32_16x16x4_f32(
      /*neg_a=*/false, a, /*neg_b=*/false, b,
      /*c_mod=*/(short)0, c, /*reuse_a=*/false, /*reuse_b=*/false);
}

__device__ __forceinline__ float bperm(float v, int srclane) {
  return __int_as_float(__builtin_amdgcn_ds_bpermute(srclane << 2, __float_as_int(v)));
}

// D-layout tile t holds rows r (r<8 in lanes 0-15, r>=8 in lanes 16-31) at vgpr r&7.
// Produce B-layout column: lanes<16 need row rlo, lanes>=16 need row rlo+2.
// rlo in {0,1,4,5,8,9,12,13}; exactly one half needs a cross-lane fetch.
__device__ __forceinline__ float makeB(const v8f& t, int rlo, int lane) {
  const int rup = rlo + 2;
  if ((rlo & 8) == 0) {
    float shuf = bperm(t[rup & 7], lane & 15);
    return (lane < 16) ? t[rlo & 7] : shuf;
  } else {
    float shuf = bperm(t[rlo & 7], (lane & 15) + 16);
    return (lane < 16) ? shuf : t[rup & 7];
  }
}

// snntorch Leaky (subtract reset, beta=0.9, thr=1):
//   reset_t = H(m_{t-1} - 1) == spike_{t-1}  ->  reuse s as both
//   m = 0.9*m + cur - s_prev ;  s = H(m - 1)
__device__ __forceinline__ void lif(v8f& m, const v8f& cur, v8f& s) {
#pragma unroll
  for (int i = 0; i < 8; ++i) {
    float mn = __builtin_fmaf(0.9f, m[i], cur[i]) - s[i];
    m[i] = mn;
    s[i] = (mn > 1.0f) ? 1.0f : 0.0f;
  }
}

__device__ __forceinline__ void nst4(float* p, float a, float b, float c, float d) {
  v4f v;
  v[0] = a; v[1] = b; v[2] = c; v[3] = d;
  __builtin_nontemporal_store(v, (v4f*)p);
}

__device__ __forceinline__ v2f ld2nt(const float* p) {
  return __builtin_nontemporal_load((const v2f*)p);
}

__global__ __launch_bounds__(32) void snn_lif3_kernel(
    const float* __restrict__ x,
    const float* __restrict__ W1, const float* __restrict__ b1,
    const float* __restrict__ W2, const float* __restrict__ b2,
    const float* __restrict__ W3, const float* __restrict__ b3,
    float* __restrict__ out) {
  const int lane = threadIdx.x;   // wave32
  const int n = lane & 15;        // batch-in-tile (B/D column, A row)
  const int h = lane >> 4;        // lane half
  const int b0 = blockIdx.x * 16; // batch tile base

  // Output regions (return order: mem3, spk3, spk1, spk2), all [T,B,*] f32
  float* mem3 = out;
  float* spk3 = out + (size_t)TT * BB * 4;
  float* spk1 = out + (size_t)TT * BB * 8;
  float* spk2 = out + (size_t)TT * BB * 36;

  // ---- Preload weights into WMMA A-layout (A[M=16][K=4] chunk: lane L holds
  // M=L&15, K = 4c + 2h + j at vgpr j). Held in VGPRs for all 1000 steps. ----
  v2f a1[2][2]; // W1 [28x8] -> 2 M-tiles x 2 K-chunks
#pragma unroll
  for (int tl = 0; tl < 2; ++tl)
#pragma unroll
    for (int c = 0; c < 2; ++c)
#pragma unroll
      for (int j = 0; j < 2; ++j) {
        int m = 16 * tl + n, k = 4 * c + 2 * h + j;
        a1[tl][c][j] = (m < 28) ? W1[m * 8 + k] : 0.0f;
      }
  v2f a2[7]; // W2 [8x28] -> 1 M-tile x 7 K-chunks
#pragma unroll
  for (int c = 0; c < 7; ++c)
#pragma unroll
    for (int j = 0; j < 2; ++j) {
      int k = 4 * c + 2 * h + j;
      a2[c][j] = (n < 8) ? W2[n * 28 + k] : 0.0f;
    }
  v2f a3[2]; // W3 [4x8] -> 1 M-tile x 2 K-chunks
#pragma unroll
  for (int c = 0; c < 2; ++c)
#pragma unroll
    for (int j = 0; j < 2; ++j) {
      int k = 4 * c + 2 * h + j;
      a3[c][j] = (n < 4) ? W3[n * 8 + k] : 0.0f;
    }

  // Biases pre-splatted into C/D layout: vgpr v, row m = v + 8h (+16 for tile1)
  v8f bia10, bia11, bia2, bia3;
#pragma unroll
  for (int v = 0; v < 8; ++v) {
    int m = v + 8 * h;
    bia10[v] = b1[m];
    bia11[v] = (16 + m < 28) ? b1[16 + m] : 0.0f;
    bia2[v]  = (m < 8) ? b2[m] : 0.0f;
    bia3[v]  = (m < 4) ? b3[m] : 0.0f;
  }

  // Membrane + spike state in D-layout registers (rows = neurons, cols = batch).
  // Spikes double as next step's reset term (H(m_prev - 1)); init 0 since m=0.
  v8f m1t0 = {0,0,0,0,0,0,0,0}, m1t1 = {0,0,0,0,0,0,0,0};
  v8f m2   = {0,0,0,0,0,0,0,0}, m3   = {0,0,0,0,0,0,0,0};
  v8f s1t0 = {0,0,0,0,0,0,0,0}, s1t1 = {0,0,0,0,0,0,0,0};
  v8f s2t  = {0,0,0,0,0,0,0,0}, s3t  = {0,0,0,0,0,0,0,0};

  // Per-lane streaming pointers
  const float* px = x    + (size_t)(b0 + n) * 8;
  float* pm3 = mem3 + (size_t)(b0 + n) * 4;
  float* ps3 = spk3 + (size_t)(b0 + n) * 4;
  float* ps1 = spk1 + (size_t)(b0 + n) * 28;
  float* ps2 = spk2 + (size_t)(b0 + n) * 8;

  // Software pipeline: x_t loads issued one step ahead so they overlap compute.
  // B-layout: vgpr j of chunk c holds k = 4c + 2h + j -> two b64 NT loads.
  v2f q0 = ld2nt(px + 2 * h);
  v2f q1 = ld2nt(px + 2 * h + 4);

#pragma unroll 1
  for (int t = 0; t < TT; ++t) {
    v2f xb0 = q0, xb1 = q1;
    const float* pnx = px + ((t < TT - 1) ? BB * 8 : 0);
    q0 = ld2nt(pnx + 2 * h);
    q1 = ld2nt(pnx + 2 * h + 4);
    px = pnx;

    // ---- layer 1: cur1^T[28x16] = W1 @ x^T (+b1), 4 WMMAs ----
    v8f c0 = bia10, c1 = bia11;
    c0 = wmma4(a1[0][0], xb0, c0);
    c0 = wmma4(a1[0][1], xb1, c0);
    c1 = wmma4(a1[1][0], xb0, c1);
    c1 = wmma4(a1[1][1], xb1, c1);

    lif(m1t0, c0, s1t0);
    lif(m1t1, c1, s1t1);

    // ---- layer 2: cur2^T[8x16] = W2 @ s1^T (+b2), 7 WMMAs; B via bpermute ----
    v8f c2 = bia2;
#pragma unroll
    for (int c = 0; c < 7; ++c) {
      const v8f& src = (c < 4) ? s1t0 : s1t1;
      v2f bv;
      bv[0] = makeB(src, (4 * c) & 15, lane);
      bv[1] = makeB(src, (4 * c + 1) & 15, lane);
      c2 = wmma4(a2[c], bv, c2);
    }
    lif(m2, c2, s2t);

    // ---- layer 3: cur3^T[4x16] = W3 @ s2^T (+b3), 2 WMMAs ----
    v8f c3 = bia3;
    {
      v2f bv;
      bv[0] = makeB(s2t, 0, lane);
      bv[1] = makeB(s2t, 1, lane);
      c3 = wmma4(a3[0], bv, c3);
      bv[0] = makeB(s2t, 4, lane);
      bv[1] = makeB(s2t, 5, lane);
      c3 = wmma4(a3[1], bv, c3);
    }
    lif(m3, c3, s3t);

    // ---- all stores grouped at step end (off the dependency chain) ----
    nst4(ps1 + 8 * h,      s1t0[0], s1t0[1], s1t0[2], s1t0[3]);
    nst4(ps1 + 8 * h + 4,  s1t0[4], s1t0[5], s1t0[6], s1t0[7]);
    nst4(ps1 + 16 + 8 * h, s1t1[0], s1t1[1], s1t1[2], s1t1[3]); // h=0: m16-19, h=1: m24-27
    if (h == 0) {
      nst4(ps1 + 20, s1t1[4], s1t1[5], s1t1[6], s1t1[7]);       // m20-23
      nst4(ps2,      s2t[0],  s2t[1],  s2t[2],  s2t[3]);
      nst4(ps2 + 4,  s2t[4],  s2t[5],  s2t[6],  s2t[7]);
      nst4(pm3,      m3[0],   m3[1],   m3[2],   m3[3]);
      nst4(ps3,      s3t[0],  s3t[1],  s3t[2],  s3t[3]);
    }

    pm3 += BB * 4;
    ps3 += BB * 4;
    ps1 += BB * 28;
    ps2 += BB * 8;
  }
}

extern "C" void kernel_launch(void* const* d_in, const int* in_sizes, int n_in,
                              void* d_out, int out_size, void* d_ws, size_t ws_size,
                              hipStream_t stream) {
  const float* x  = (const float*)d_in[0];
  const float* W1 = (const float*)d_in[1];
  const float* b1 = (const float*)d_in[2];
  const float* W2 = (const float*)d_in[3];
  const float* b2 = (const float*)d_in[4];
  const float* W3 = (const float*)d_in[5];
  const float* b3 = (const float*)d_in[6];
  (void)in_sizes; (void)n_in; (void)out_size; (void)d_ws; (void)ws_size;
  snn_lif3_kernel<<<BB / 16, 32, 0, stream>>>(x, W1, b1, W2, b2, W3, b3, (float*)d_out);
}